// BioTripletLoss_78520592105502
// MI455X (gfx1250) — compile-verified
//
#include <hip/hip_runtime.h>
#include <cstdint>

typedef __attribute__((ext_vector_type(2))) float v2f;
typedef __attribute__((ext_vector_type(8))) float v8f;

#define WAVES_PER_BLOCK 4
#define ROWS_PER_WAVE   16
#define TK              64          // columns staged per stage
#define LDS_STRIDE      68          // padded row stride in floats (68%64 banks = 4 -> conflict-free b64 reads)
#define TILE_FLOATS     (ROWS_PER_WAVE * LDS_STRIDE)   // 1088
#define BUF_FLOATS      (4 * TILE_FLOATS)              // h,r,t,tn tiles

// low 32 bits of a generic pointer into LDS == LDS byte offset
__device__ __forceinline__ unsigned lds_off(const void* p) {
  return (unsigned)(uintptr_t)p;
}

// CDNA5 async copy: 16 bytes per lane, global -> LDS, tracked by ASYNCcnt
__device__ __forceinline__ void async_copy16(unsigned lds_addr, const float* gsrc) {
  asm volatile("global_load_async_to_lds_b128 %0, %1, off"
               :: "v"(lds_addr), "v"(gsrc) : "memory");
}
__device__ __forceinline__ void wait_async_le0()  { asm volatile("s_wait_asynccnt 0x0"  ::: "memory"); }
__device__ __forceinline__ void wait_async_le32() { asm volatile("s_wait_asynccnt 0x20" ::: "memory"); }

__global__ __launch_bounds__(WAVES_PER_BLOCK * 32)
void BioTripletLoss_rows_kernel(const float* __restrict__ h,
                                const float* __restrict__ t,
                                const float* __restrict__ r,
                                const int*   __restrict__ rel,
                                const int*   __restrict__ neg,
                                float*       __restrict__ row_loss,
                                int Bn, int Dn) {
  // double-buffered: [wave][buf][4 tiles]
  __shared__ float smem[WAVES_PER_BLOCK * 2 * BUF_FLOATS];

  const int tid  = (int)threadIdx.x;
  const int wave = tid >> 5;
  const int lane = tid & 31;
  const int rowbase = ((int)blockIdx.x * WAVES_PER_BLOCK + wave) * ROWS_PER_WAVE;

  float* buf0 = &smem[(wave * 2 + 0) * BUF_FLOATS];
  float* buf1 = &smem[(wave * 2 + 1) * BUF_FLOATS];

  const int rl = lane >> 4;   // staging: row within pair (0/1)
  const int cl = lane & 15;   // staging: 16-byte column group

  // WMMA feed mapping: lane holds A[m = lane%16][k + 2*(lane/16) + {0,1}]
  const int rowA  = lane & 15;
  const int khalf = (lane >> 4) * 2;

  // hoist gathered-row indices (loop-invariant across stages)
  int nrow[8];
#pragma unroll
  for (int p = 0; p < 8; ++p) nrow[p] = neg[rowbase + 2 * p + rl];

  v8f cpos = {};  // Gram accumulator: diag = sum of squares (pos)
  v8f cneg = {};  // Gram accumulator: diag = sum of squares (neg)

  // issue 32 async b128 copies staging 16 rows x 64 cols of h,r,t,t[neg]
  auto issue_stage = [&](float* buf, int k0) {
    float* th  = buf + 0 * TILE_FLOATS;
    float* tr_ = buf + 1 * TILE_FLOATS;
    float* tt  = buf + 2 * TILE_FLOATS;
    float* tn  = buf + 3 * TILE_FLOATS;
#pragma unroll
    for (int p = 0; p < 8; ++p) {
      const int tr = 2 * p + rl;
      const size_t gr = (size_t)(rowbase + tr) * (size_t)Dn + (size_t)k0 + (size_t)cl * 4;
      const unsigned lo = (unsigned)(tr * LDS_STRIDE + cl * 4) * 4u;
      async_copy16(lds_off(th)  + lo, h + gr);
      async_copy16(lds_off(tr_) + lo, r + gr);
      async_copy16(lds_off(tt)  + lo, t + gr);
      const size_t gn = (size_t)nrow[p] * (size_t)Dn + (size_t)k0 + (size_t)cl * 4;
      async_copy16(lds_off(tn)  + lo, t + gn);
    }
  };

  // consume one resident stage: 16 K-chunks of 4, two Gram WMMAs per chunk
  auto consume_stage = [&](const float* buf) {
    const float* th = buf + 0 * TILE_FLOATS;
    const float* tr_ = buf + 1 * TILE_FLOATS;
    const float* tt = buf + 2 * TILE_FLOATS;
    const float* tn = buf + 3 * TILE_FLOATS;
#pragma unroll
    for (int kk = 0; kk < TK; kk += 4) {
      const int fo = rowA * LDS_STRIDE + kk + khalf;
      v2f hh = *(const v2f*)&th[fo];
      v2f rr = *(const v2f*)&tr_[fo];
      v2f t0 = *(const v2f*)&tt[fo];
      v2f n0 = *(const v2f*)&tn[fo];
      v2f hr = hh + rr;
      v2f dp = hr - t0;
      v2f dn = hr - n0;
      // A and A^T share one register image for 16x16x4 f32: C[m][m] += sum_k dp[m][k]^2
      cpos = __builtin_amdgcn_wmma_f32_16x16x4_f32(false, dp, false, dp, (short)0, cpos, false, false);
      cneg = __builtin_amdgcn_wmma_f32_16x16x4_f32(false, dn, false, dn, (short)0, cneg, false, false);
    }
  };

  // software pipeline: prefetch next stage, wait for previous (in-order async completion)
  issue_stage(buf0, 0);
  for (int k0 = 0; k0 < Dn; k0 += 2 * TK) {
    if (k0 + TK < Dn) { issue_stage(buf1, k0 + TK); wait_async_le32(); }
    else              { wait_async_le0(); }
    consume_stage(buf0);
    if (k0 + TK < Dn) {
      if (k0 + 2 * TK < Dn) { issue_stage(buf0, k0 + 2 * TK); wait_async_le32(); }
      else                  { wait_async_le0(); }
      consume_stage(buf1);
    }
  }

  // extract Gram diagonals: row m lives at (vgpr m&7, lane m<8 ? m : m+16)
  float psum = 0.f, nsum = 0.f;
#pragma unroll
  for (int m = 0; m < 16; ++m) {
    const int sl = (m < 8) ? m : (m + 16);
    const float pv = __shfl(cpos[m & 7], sl, 32);
    const float nv = __shfl(cneg[m & 7], sl, 32);
    if (lane == m) { psum = pv; nsum = nv; }
  }

  if (lane < 16) {
    const int row = rowbase + lane;
    const float pos = sqrtf(psum);
    const float nd  = sqrtf(nsum);
    float loss;
    if (rel[row] == 1) {
      loss = fmaxf(0.6f - pos, 0.f) + 0.5f * expf(-pos);           // relu(0.6-pos)+0.5*exp(-pos)
    } else {
      loss = fmaxf(pos - nd + 0.3f, 0.f) + 0.3f * fmaxf(0.1f - pos, 0.f);
    }
    row_loss[row] = loss;
  }
}

__global__ __launch_bounds__(256)
void BioTripletLoss_reduce_kernel(const float* __restrict__ vals,
                                  float* __restrict__ out, int n) {
  __shared__ float red[256];
  float s = 0.f;
  for (int i = (int)threadIdx.x; i < n; i += 256) s += vals[i];
  red[threadIdx.x] = s;
  __syncthreads();
  for (int off = 128; off > 0; off >>= 1) {
    if ((int)threadIdx.x < off) red[threadIdx.x] += red[threadIdx.x + off];
    __syncthreads();
  }
  if (threadIdx.x == 0) out[0] = red[0] / (float)n;
}

extern "C" void kernel_launch(void* const* d_in, const int* in_sizes, int n_in,
                              void* d_out, int out_size, void* d_ws, size_t ws_size,
                              hipStream_t stream) {
  const float* h  = (const float*)d_in[0];
  const float* t  = (const float*)d_in[1];
  const float* r  = (const float*)d_in[2];
  const int*  rel = (const int*)d_in[3];
  const int*  neg = (const int*)d_in[4];

  const int Bn = in_sizes[3];            // 16384
  const int Dn = in_sizes[0] / Bn;       // 1024

  float* row_loss = (float*)d_ws;        // Bn floats of scratch

  const int rows_per_block = WAVES_PER_BLOCK * ROWS_PER_WAVE;   // 64
  const int grid = (Bn + rows_per_block - 1) / rows_per_block;  // 256 blocks

  BioTripletLoss_rows_kernel<<<grid, WAVES_PER_BLOCK * 32, 0, stream>>>(
      h, t, r, rel, neg, row_loss, Bn, Dn);
  BioTripletLoss_reduce_kernel<<<1, 256, 0, stream>>>(row_loss, (float*)d_out, Bn);
}